// Loss_Synonymy_44487271252808
// MI455X (gfx1250) — compile-verified
//
#include <hip/hip_runtime.h>
#include <math.h>

// ---------------------------------------------------------------------------
// Loss_Synonymy on MI455X (gfx1250, wave32)
//
// out = (1/B) * sum_i  relu( score[i] >= 0.8 ? 1 - tanh(||s1_i - s2_i||_2)
//                                            : 1 + tanh(||s1_i - s2_i||_2) )
//
// Memory-bound: 512 MB streamed once -> ~22 us floor @ 23.3 TB/s.
// Row sum-of-squares is done on the WMMA unit: D = A x Ones + C with
// V_WMMA_F32_16X16X4_F32, one 16-row tile per wave per WMMA chain.
// Two-stage deterministic reduction (no float atomics).
// ---------------------------------------------------------------------------

typedef float v2f __attribute__((ext_vector_type(2)));
typedef float v8f __attribute__((ext_vector_type(8)));

#define DCOLS   256
#define THRESH  0.8f
#define BLOCK   256                    // 8 waves per block (wave32)
#define WPB     (BLOCK / 32)
#define NBLOCKS 1024                   // 8192 waves in flight; 4 KB of partials

__global__ __launch_bounds__(BLOCK)
void synonymy_partial_kernel(const float* __restrict__ s1,
                             const float* __restrict__ s2,
                             const float* __restrict__ score,
                             float* __restrict__ partials,
                             int nrows)
{
    const int lane = threadIdx.x & 31;
    const int wave = threadIdx.x >> 5;
    const int half = lane >> 4;        // 0: K-cols {0,1}; 1: K-cols {2,3}
    const int m    = lane & 15;        // row-in-tile handled by this lane

    const int nTiles     = nrows >> 4;                 // 16 rows per tile
    const int wavesTotal = gridDim.x * WPB;
    const int waveGid    = blockIdx.x * WPB + wave;

    // B-matrix of all ones: layout-invariant, so D[m][n] = sum_k A[m][k] + C.
    v2f bones; bones.x = 1.0f; bones.y = 1.0f;

    float waveAcc = 0.0f;

    for (int tile = waveGid; tile < nTiles; tile += wavesTotal) {
        const int rowBase = tile << 4;
        // Lane's fixed row; each K-chunk loads an aligned float2 per input.
        const float* p1 = s1 + (size_t)(rowBase + m) * DCOLS + half * 2;
        const float* p2 = s2 + (size_t)(rowBase + m) * DCOLS + half * 2;

        v8f acc = {0.f, 0.f, 0.f, 0.f, 0.f, 0.f, 0.f, 0.f};

        // 64 chunks of K=4 cover all 256 columns; every element read once.
        #pragma unroll 4
        for (int kb = 0; kb < 64; ++kb) {
            v2f x1 = *(const v2f*)(p1 + kb * 4);
            v2f x2 = *(const v2f*)(p2 + kb * 4);
            float d0 = x1.x - x2.x;
            float d1 = x1.y - x2.y;
            v2f a; a.x = d0 * d0; a.y = d1 * d1;   // A-matrix 16x4 f32 layout
            // D = A x Ones + C : accumulates per-row sum of squares.
            acc = __builtin_amdgcn_wmma_f32_16x16x4_f32(
                false, a, false, bones, (short)0, acc, false, false);
        }

        // C/D layout: VGPR r = rowsum(rowBase + r)   in lanes 0..15,
        //                      rowsum(rowBase + 8+r) in lanes 16..31.
        // All 16 lanes of a half hold identical values -> compute redundantly.
        const int rb = rowBase + half * 8;
        float e = 0.0f;
        #pragma unroll
        for (int r = 0; r < 8; ++r) {
            float d  = sqrtf(acc[r]);
            float t  = tanhf(d);
            float sc = score[rb + r];
            float v  = (sc >= THRESH) ? (1.0f - t) : (1.0f + t);
            e += fmaxf(v, 0.0f);
        }
        // Merge the two halves (each half's e is uniform within the half).
        e += __shfl_xor(e, 16, 32);
        waveAcc += e;
    }

    // Deterministic in-block reduction: 8 wave partials summed in fixed order.
    __shared__ float sdata[WPB];
    if (lane == 0) sdata[wave] = waveAcc;
    __syncthreads();
    if (threadIdx.x == 0) {
        float s = 0.0f;
        #pragma unroll
        for (int w = 0; w < WPB; ++w) s += sdata[w];
        partials[blockIdx.x] = s;
    }
}

__global__ __launch_bounds__(256)
void synonymy_final_kernel(const float* __restrict__ partials,
                           float* __restrict__ out,
                           int nPartials, float invN)
{
    __shared__ float sdata[256];
    float s = 0.0f;
    for (int i = threadIdx.x; i < nPartials; i += 256) s += partials[i];
    sdata[threadIdx.x] = s;
    __syncthreads();
    #pragma unroll
    for (int off = 128; off > 0; off >>= 1) {
        if (threadIdx.x < off) sdata[threadIdx.x] += sdata[threadIdx.x + off];
        __syncthreads();
    }
    if (threadIdx.x == 0) out[0] = sdata[0] * invN;
}

extern "C" void kernel_launch(void* const* d_in, const int* in_sizes, int n_in,
                              void* d_out, int out_size, void* d_ws, size_t ws_size,
                              hipStream_t stream)
{
    const float* s1    = (const float*)d_in[0];   // S1_out  [B, 256] f32
    const float* s2    = (const float*)d_in[1];   // S2_out  [B, 256] f32
    const float* score = (const float*)d_in[2];   // synonymy_score [B] f32
    const int nrows    = in_sizes[2];             // B (= 262144, multiple of 16)

    float* partials = (float*)d_ws;               // NBLOCKS * 4 bytes scratch

    synonymy_partial_kernel<<<NBLOCKS, BLOCK, 0, stream>>>(
        s1, s2, score, partials, nrows);
    synonymy_final_kernel<<<1, 256, 0, stream>>>(
        partials, (float*)d_out, NBLOCKS, 1.0f / (float)nrows);
}